// GraphSageConv_50105088475799
// MI455X (gfx1250) — compile-verified
//
#include <hip/hip_runtime.h>

typedef __attribute__((ext_vector_type(2))) float v2f;
typedef __attribute__((ext_vector_type(8))) float v8f;

#define F_DIM 128
#define C_DIM 256
#define K_DIM 256   // 2*F
#define AS 68       // LDS stride for A tile (64 + 4 pad -> conflict-free b64 frag reads)
#define BS 80       // LDS stride for B tile (64 + 16 pad -> disjoint half-wave banks)

// ---------------- Kernel 1: segment-mean aggregation + concat [X || mean] ----
// One wave per node (edge_row is sorted -> binary search, no atomics).
__global__ __launch_bounds__(256) void sage_aggregate(
    const float* __restrict__ feat, const int* __restrict__ erow,
    const int* __restrict__ ecol, float* __restrict__ Xc, int N, int E)
{
    const int lane = threadIdx.x;                 // 0..31, owns 4 of 128 features
    const int node = blockIdx.x * 8 + threadIdx.y;
    if (node >= N) return;

    // lower_bound(erow, node)
    int lo = 0, hi = E;
    while (lo < hi) { int mid = (lo + hi) >> 1; if (erow[mid] < node) lo = mid + 1; else hi = mid; }
    const int start = lo;
    // upper_bound(erow, node)
    hi = E;
    while (lo < hi) { int mid = (lo + hi) >> 1; if (erow[mid] <= node) lo = mid + 1; else hi = mid; }
    const int end = lo;

    float a0 = 0.f, a1 = 0.f, a2 = 0.f, a3 = 0.f;
    for (int e = start; e < end; ++e) {
        const int c = ecol[e];
        const float4 v = *(const float4*)(feat + (size_t)c * F_DIM + lane * 4);
        a0 += v.x; a1 += v.y; a2 += v.z; a3 += v.w;
    }
    const float inv = 1.0f / fmaxf((float)(end - start), 1.0f);
    const float4 self = *(const float4*)(feat + (size_t)node * F_DIM + lane * 4);
    float* dst = Xc + (size_t)node * K_DIM;
    *(float4*)(dst + lane * 4) = self;
    float4 mean; mean.x = a0 * inv; mean.y = a1 * inv; mean.z = a2 * inv; mean.w = a3 * inv;
    *(float4*)(dst + F_DIM + lane * 4) = mean;
}

// ---------------- Kernel 2: [N,256] x [256,256] fp32 WMMA GEMM + bias -------
__global__ __launch_bounds__(512) void sage_gemm(
    const float* __restrict__ Xc, const float* __restrict__ W,
    const float* __restrict__ bias, float* __restrict__ out, int N)
{
    __shared__ float Alds[64 * AS];
    __shared__ float Blds[64 * BS];

    const int tid  = threadIdx.x;
    const int lane = tid & 31;
    const int wave = tid >> 5;           // 0..15
    const int mt   = wave >> 2;          // 0..3  (M subtile)
    const int nt   = wave & 3;           // 0..3  (N subtile)
    const int m0   = blockIdx.x * 64;
    const int n0   = blockIdx.y * 64;

    const int l15  = lane & 15;
    const int kb   = (lane >> 4) << 1;   // 0 or 2 (K split across half-waves)
    const int arow = mt * 16 + l15;      // A fragment LDS row
    const int ncol = nt * 16 + l15;      // B fragment LDS col

    v8f acc = {};
    for (int kc = 0; kc < K_DIM; kc += 64) {
        // stage A: 64 rows x 64 K, zero-pad rows past N
        for (int i = tid; i < 1024; i += 512) {
            const int r = i >> 4, c4 = (i & 15) << 2;
            const int gr = m0 + r;
            float4 v;
            if (gr < N) v = *(const float4*)(Xc + (size_t)gr * K_DIM + kc + c4);
            else { v.x = v.y = v.z = v.w = 0.0f; }
            *(float4*)(Alds + r * AS + c4) = v;
        }
        // stage B: 64 K x 64 N
        for (int i = tid; i < 1024; i += 512) {
            const int r = i >> 4, c4 = (i & 15) << 2;
            const float4 v = *(const float4*)(W + (size_t)(kc + r) * C_DIM + n0 + c4);
            *(float4*)(Blds + r * BS + c4) = v;
        }
        __syncthreads();

        #pragma unroll
        for (int k0 = 0; k0 < 64; k0 += 4) {
            const v2f a = *(const v2f*)(Alds + arow * AS + k0 + kb);
            v2f b;
            b.x = Blds[(k0 + kb    ) * BS + ncol];
            b.y = Blds[(k0 + kb + 1) * BS + ncol];
            // 8 args: (neg_a, A, neg_b, B, c_mod, C, reuse_a, reuse_b)
            acc = __builtin_amdgcn_wmma_f32_16x16x4_f32(
                false, a, false, b, (short)0, acc, false, false);
        }
        __syncthreads();
    }

    const float bv    = bias[n0 + ncol];
    const int   mbase = m0 + mt * 16 + (lane >> 4) * 8;
    #pragma unroll
    for (int v = 0; v < 8; ++v) {
        const int gr = mbase + v;
        if (gr < N) out[(size_t)gr * C_DIM + n0 + ncol] = acc[v] + bv;
    }
}

// ---------------- Kernel 3: row-wise L2 normalize (in place) ----------------
__global__ __launch_bounds__(256) void sage_l2norm(float* __restrict__ out, int N)
{
    const int lane = threadIdx.x & 31;
    const int wave = threadIdx.x >> 5;
    const int row  = blockIdx.x * 8 + wave;
    if (row >= N) return;
    float* p = out + (size_t)row * C_DIM;
    float4 v0 = *(const float4*)(p + lane * 4);
    float4 v1 = *(const float4*)(p + 128 + lane * 4);
    float s = v0.x*v0.x + v0.y*v0.y + v0.z*v0.z + v0.w*v0.w
            + v1.x*v1.x + v1.y*v1.y + v1.z*v1.z + v1.w*v1.w;
    #pragma unroll
    for (int off = 16; off > 0; off >>= 1) s += __shfl_xor(s, off, 32);
    const float r = rsqrtf(fmaxf(s, 1e-12f));
    v0.x *= r; v0.y *= r; v0.z *= r; v0.w *= r;
    v1.x *= r; v1.y *= r; v1.z *= r; v1.w *= r;
    *(float4*)(p + lane * 4) = v0;
    *(float4*)(p + 128 + lane * 4) = v1;
}

extern "C" void kernel_launch(void* const* d_in, const int* in_sizes, int n_in,
                              void* d_out, int out_size, void* d_ws, size_t ws_size,
                              hipStream_t stream) {
    const float* feat = (const float*)d_in[0];
    const int*   erow = (const int*)d_in[1];
    const int*   ecol = (const int*)d_in[2];
    const float* W    = (const float*)d_in[3];
    const float* bias = (const float*)d_in[4];
    float*       out  = (float*)d_out;
    const int N = in_sizes[0] / F_DIM;
    const int E = in_sizes[1];
    float* Xc = (float*)d_ws;   // [N, 256] concat workspace (~51 MB)

    dim3 b1(32, 8);
    sage_aggregate<<<(N + 7) / 8, b1, 0, stream>>>(feat, erow, ecol, Xc, N, E);

    dim3 g2((N + 63) / 64, C_DIM / 64);
    sage_gemm<<<g2, dim3(512), 0, stream>>>(Xc, W, bias, out, N);

    sage_l2norm<<<(N + 7) / 8, dim3(256), 0, stream>>>(out, N);
}